// Net_12438225289954
// MI455X (gfx1250) — compile-verified
//
#include <hip/hip_runtime.h>
#include <hip/hip_bf16.h>

typedef __attribute__((ext_vector_type(2))) float v2f;
typedef __attribute__((ext_vector_type(8))) float v8f;

#define D 128
#define ROWS_PER_WG 128  // 8 row-tiles of 16 per workgroup

// ---------------------------------------------------------------------------
// z = src (float4-vectorized grid-stride copy)
// ---------------------------------------------------------------------------
__global__ void copy4_kernel(const float* __restrict__ src, float* __restrict__ dst,
                             long n4) {
  long i = (long)blockIdx.x * blockDim.x + threadIdx.x;
  long stride = (long)gridDim.x * blockDim.x;
  const float4* s = (const float4*)src;
  float4* d = (float4*)dst;
  for (; i < n4; i += stride) d[i] = s[i];
}

// ---------------------------------------------------------------------------
// z[dst[e]] += h[src[e]]  -- one wave32 per edge, float4 per lane,
// 4x global_atomic_add_f32 (no-return).  Edge indices are int64 per reference.
// ---------------------------------------------------------------------------
__global__ void scatter_add_kernel(const float* __restrict__ h,
                                   const long long* __restrict__ ei,
                                   float* __restrict__ z, int nE) {
  int e = blockIdx.x * (blockDim.x >> 5) + (threadIdx.x >> 5);
  int lane = threadIdx.x & 31;
  if (e >= nE) return;
  int s = (int)ei[e];        // row 0 of edge_index: src
  int d = (int)ei[nE + e];   // row 1 of edge_index: dst
  float4 v = ((const float4*)(h + (size_t)s * D))[lane];
  float* zd = z + (size_t)d * D + lane * 4;
  atomicAdd(zd + 0, v.x);
  atomicAdd(zd + 1, v.y);
  atomicAdd(zd + 2, v.z);
  atomicAdd(zd + 3, v.w);
}

// ---------------------------------------------------------------------------
// Out = relu(Z @ W^T + bias)   Z:[nRows,128]  W:[128,128] (torch [out,in])
// 256 threads = 8 waves; wave w owns output columns [16w, 16w+16).
//
// B fragment (4x16 tiles of W^T for this wave's columns) is loop-invariant:
// hoisted once into 64 VGPRs (32 steps x v2f).  Per 16-row tile: 32
// independent global_load_b64 for A feed 32 chained v_wmma_f32_16x16x4_f32.
// No LDS, no barriers; EXEC stays all-1s (A row index clamped, stores masked).
//
// Lane layouts per ISA 7.12.2 (wave32, f32):
//   A(16x4):  a[j] = A[lm][k + 2*half + j]
//   B(4x16):  b[j] = B[k + 2*half + j][n0 + lm] = W[n0+lm][k + 2*half + j]
//   C/D:      c[v] -> row (v + 8*half), col lm  of the 16x16 tile
// ---------------------------------------------------------------------------
__global__ void gin_gemm_relu_kernel(const float* __restrict__ Z,
                                     const float* __restrict__ W,
                                     const float* __restrict__ bias,
                                     float* __restrict__ Out, int nRows) {
  const int tid  = threadIdx.x;
  const int wave = tid >> 5;
  const int lane = tid & 31;
  const int half = lane >> 4;   // 0: lanes 0-15, 1: lanes 16-31
  const int lm   = lane & 15;
  const int n0   = wave * 16;   // this wave's output-column tile

  // Hoist B into registers: lane lm holds row (n0+lm) of W, elements
  // {k+2*half, k+2*half+1} for each of the 32 K-steps.  64 VGPRs total.
  v2f b[32];
  {
    const float* wrow = W + (size_t)(n0 + lm) * D + 2 * half;
#pragma unroll
    for (int s = 0; s < 32; ++s) b[s] = *(const v2f*)(wrow + 4 * s);
  }
  const float bv = bias[n0 + lm];

  int row0 = blockIdx.x * ROWS_PER_WG;
  const int rowEnd = (row0 + ROWS_PER_WG < nRows) ? row0 + ROWS_PER_WG : nRows;
  for (; row0 < rowEnd; row0 += 16) {
    int ar = row0 + lm;
    if (ar > nRows - 1) ar = nRows - 1;  // clamp A reads; outputs masked below
    const float* zrow = Z + (size_t)ar * D + 2 * half;

    v8f c = {};
#pragma unroll
    for (int s = 0; s < 32; ++s) {
      v2f a = *(const v2f*)(zrow + 4 * s);
      c = __builtin_amdgcn_wmma_f32_16x16x4_f32(
          /*neg_a=*/false, a, /*neg_b=*/false, b[s],
          /*c_mod=*/(short)0, c, /*reuse_a=*/false, /*reuse_b=*/false);
    }

#pragma unroll
    for (int v = 0; v < 8; ++v) {
      int gr = row0 + v + 8 * half;
      if (gr < nRows) Out[(size_t)gr * D + n0 + lm] = fmaxf(c[v] + bv, 0.0f);
    }
  }
}

// ---------------------------------------------------------------------------
// out[n] = dot(h[n,:], Wf) + bf   -- one wave32 per node
// ---------------------------------------------------------------------------
__global__ void gin_final_kernel(const float* __restrict__ h,
                                 const float* __restrict__ Wf,
                                 const float* __restrict__ bf,
                                 float* __restrict__ out, int n) {
  int node = blockIdx.x * (blockDim.x >> 5) + (threadIdx.x >> 5);
  int lane = threadIdx.x & 31;
  if (node >= n) return;
  float4 w = ((const float4*)Wf)[lane];
  float4 v = ((const float4*)(h + (size_t)node * D))[lane];
  float s = v.x * w.x + v.y * w.y + v.z * w.z + v.w * w.w;
#pragma unroll
  for (int off = 16; off > 0; off >>= 1) s += __shfl_down(s, off, 32);
  if (lane == 0) out[node] = s + bf[0];
}

// ---------------------------------------------------------------------------
extern "C" void kernel_launch(void* const* d_in, const int* in_sizes, int n_in,
                              void* d_out, int out_size, void* d_ws, size_t ws_size,
                              hipStream_t stream) {
  const float* x      = (const float*)d_in[0];
  const long long* ei = (const long long*)d_in[1];
  const float* W1     = (const float*)d_in[2];
  const float* b1     = (const float*)d_in[3];
  const float* W2     = (const float*)d_in[4];
  const float* b2     = (const float*)d_in[5];
  const float* Wf     = (const float*)d_in[6];
  const float* bf     = (const float*)d_in[7];
  float* out = (float*)d_out;

  const int N = in_sizes[0] / D;  // 100000
  const int E = in_sizes[1] / 2;  // 600000
  const size_t Nf = (size_t)N * D;

  float* h = (float*)d_ws;  // 3 x [N,128] fp32 scratch buffers
  float* z = h + Nf;
  float* t = z + Nf;

  const dim3 blk(256);
  const int copyBlocks    = 2048;
  const int scatterBlocks = (E + 7) / 8;                       // 8 edges/block
  const int gemmBlocks    = (N + ROWS_PER_WG - 1) / ROWS_PER_WG;
  const int finalBlocks   = (N + 7) / 8;

  const float* hin = x;  // block 0 reads the original features
  for (int i = 0; i < 3; ++i) {
    const float* Wa = W1 + (size_t)i * D * D;
    const float* ba = b1 + (size_t)i * D;
    const float* Wb = W2 + (size_t)i * D * D;
    const float* bb = b2 + (size_t)i * D;

    // z = hin  (self term), then z += sum_{neighbors} hin[src]
    copy4_kernel<<<copyBlocks, blk, 0, stream>>>(hin, z, (long)(Nf / 4));
    scatter_add_kernel<<<scatterBlocks, blk, 0, stream>>>(hin, ei, z, E);
    // t = relu(z @ W1^T + b1);  h = relu(t @ W2^T + b2)
    gin_gemm_relu_kernel<<<gemmBlocks, blk, 0, stream>>>(z, Wa, ba, t, N);
    gin_gemm_relu_kernel<<<gemmBlocks, blk, 0, stream>>>(t, Wb, bb, h, N);
    hin = h;
  }
  gin_final_kernel<<<finalBlocks, blk, 0, stream>>>(h, Wf, bf, out, N);
}